// RecurrentInhibition_22608707846658
// MI455X (gfx1250) — compile-verified
//
#include <hip/hip_runtime.h>
#include <hip/hip_bf16.h>

typedef __attribute__((ext_vector_type(2))) float v2f;
typedef __attribute__((ext_vector_type(4))) float v4f;
typedef __attribute__((ext_vector_type(8))) float v8f;

#define MH_SCOPE     27
#define MH_HALF      13
#define NCH          128
#define SPATIAL      3136            // 56*56
#define NBATCH       64
#define NCOLS        (NBATCH * SPATIAL)   // 200704 = 1568*128 exactly
#define CH_STRIDE    SPATIAL
#define BATCH_STRIDE (NCH * SPATIAL)      // 401408
#define LDS_STRIDE   132             // 528B rows: 16B aligned, 132%64=4 -> conflict-free

// ---------------------------------------------------------------------------
// Kernel 1: build the 128x128 effective operator M = S_20 where
//   S_0 = C,  S_{t+1} = 0.95*S_t + 0.025*C + 0.025*(C @ S_t)
// C is the banded conv matrix of the (symmetric) mexican-hat filter.
// (C @ S)[:,j] is just the 27-tap conv of column j along the row index,
// so each of the 128 columns evolves independently -> one workgroup each.
// ---------------------------------------------------------------------------
__global__ __launch_bounds__(128) void build_M(const float* __restrict__ w,
                                               float* __restrict__ M) {
  __shared__ float wl[MH_SCOPE];
  __shared__ float sbuf[2][NCH + 2 * MH_HALF];   // zero-padded halo
  const int i = threadIdx.x;   // row index 0..127
  const int j = blockIdx.x;    // column of S

  if (i < MH_SCOPE) wl[i] = w[i];
  for (int idx = i; idx < NCH + 2 * MH_HALF; idx += 128) {
    sbuf[0][idx] = 0.f;
    sbuf[1][idx] = 0.f;
  }
  __syncthreads();

  const int d0 = i - j + MH_HALF;
  const float ci = (d0 >= 0 && d0 < MH_SCOPE) ? wl[d0] : 0.f;   // C[i,j]
  sbuf[0][MH_HALF + i] = ci;                                    // S_0 = C
  __syncthreads();

  int cur = 0;
  for (int t = 0; t < 20; ++t) {
    float conv = 0.f;
#pragma unroll
    for (int d = 0; d < MH_SCOPE; ++d)
      conv += wl[d] * sbuf[cur][i + d];        // (C @ S)[i,j]
    const float nv = 0.95f * sbuf[cur][MH_HALF + i] + 0.025f * ci + 0.025f * conv;
    sbuf[1 - cur][MH_HALF + i] = nv;           // write other buffer (safe)
    __syncthreads();                           // writes visible before next reads
    cur ^= 1;
  }
  M[i * NCH + j] = sbuf[cur][MH_HALF + i];
}

// ---------------------------------------------------------------------------
// Kernel 2: out[n,:,s] = M @ x[n,:,s] for all 200704 columns.
// 256 threads = 8 waves; M cached in LDS (padded rows). Each wave computes
// 128 rows x 16 columns via 8 f32 WMMA accumulator tiles, K-loop step 4.
// f32 WMMA fragment layouts (ISA 7.12.2):
//   A 16x4 : lane<16 -> row=lane, {K=k0,k0+1}; lane>=16 -> row=lane-16, {K=k0+2,k0+3}
//   B 4x16 : lane<16 -> col=lane, {K=k0,k0+1}; lane>=16 -> col=lane-16, {K=k0+2,k0+3}
//   D 16x16: vgpr v -> row v (lanes 0-15) / row v+8 (lanes 16-31), col=lane&15
// ---------------------------------------------------------------------------
__global__ __launch_bounds__(256) void gemm_wmma(const float* __restrict__ X,
                                                 const float* __restrict__ Mg,
                                                 float* __restrict__ Y) {
  __shared__ float lm[NCH * LDS_STRIDE];   // 67,584 B
  const int tid = threadIdx.x;

  // cooperative load of M into padded LDS, 16B vectors
  for (int it = tid; it < NCH * (NCH / 4); it += 256) {
    const int row = it >> 5;            // 0..127
    const int c4  = (it & 31) << 2;     // 0..124 step 4
    v4f v = *(const v4f*)(Mg + row * NCH + c4);
    *(v4f*)(&lm[row * LDS_STRIDE + c4]) = v;
  }
  __syncthreads();

  const int wave = tid >> 5;
  const int lane = tid & 31;
  const int cl   = lane & 15;
  const int kh   = (lane >> 4) << 1;    // 0 or 2: K offset for upper half-wave

  const int  q    = blockIdx.x * 128 + wave * 16 + cl;   // global column id
  const int  n    = q / SPATIAL;
  const int  s    = q - n * SPATIAL;
  const long base = (long)n * BATCH_STRIDE + s;          // offset at channel 0
  const float* xcol = X + base;

  v8f acc[8] = {};
  for (int k0 = 0; k0 < NCH; k0 += 4) {
    v2f b;
    b.x = xcol[(k0 + kh) * CH_STRIDE];
    b.y = xcol[(k0 + kh + 1) * CH_STRIDE];
#pragma unroll
    for (int t = 0; t < 8; ++t) {
      v2f a = *(const v2f*)(&lm[(t * 16 + cl) * LDS_STRIDE + k0 + kh]);
      acc[t] = __builtin_amdgcn_wmma_f32_16x16x4_f32(
          /*neg_a=*/false, a, /*neg_b=*/false, b,
          /*c_mod=*/(short)0, acc[t], /*reuse_a=*/false, /*reuse_b=*/false);
    }
  }

  float* ycol = Y + base;
  const int rowadd = (lane >> 4) << 3;   // 0 or 8
#pragma unroll
  for (int t = 0; t < 8; ++t) {
#pragma unroll
    for (int v = 0; v < 8; ++v) {
      ycol[(t * 16 + v + rowadd) * CH_STRIDE] = acc[t][v];
    }
  }
}

// ---------------------------------------------------------------------------
extern "C" void kernel_launch(void* const* d_in, const int* in_sizes, int n_in,
                              void* d_out, int out_size, void* d_ws, size_t ws_size,
                              hipStream_t stream) {
  const float* act = (const float*)d_in[0];   // (64,128,56,56) f32
  const float* wgt = (const float*)d_in[1];   // 27-tap mexican hat f32
  float* M   = (float*)d_ws;                  // 128*128 f32 scratch (64 KB)
  float* out = (float*)d_out;                 // (64,128,56,56) f32

  build_M<<<NCH, 128, 0, stream>>>(wgt, M);
  gemm_wmma<<<NCOLS / 128, 256, 0, stream>>>(act, M, out);
}